// JointOldCandidate_40750649704562
// MI455X (gfx1250) — compile-verified
//
#include <hip/hip_runtime.h>
#include <stdint.h>

// Problem constants from the reference: B=8, L=64, C=32.
#define BB 8
#define LL 64
#define CC 32
// Padded LDS row stride (dwords) for t2: 32 data + 2 pad, inserted by the TDM.
// Lane k reads float2 chunks at dword addr k*34 + 2h -> bank 2*(17k mod 32):
// all 32 lanes of a wave hit distinct banks => conflict-free ds_load_b64.
#define T2S 34

typedef unsigned int v4u __attribute__((ext_vector_type(4)));
typedef int          v8i __attribute__((ext_vector_type(8)));
typedef int          v4i __attribute__((ext_vector_type(4)));

__global__ __launch_bounds__(256) void jk_argmax_kernel(
    const float* __restrict__ t0,
    const float* __restrict__ t1,
    const float* __restrict__ t2,
    float* __restrict__ out)
{
    __shared__ __align__(16) float lds_t2[LL * T2S];   // 8704 B, padded rows
    __shared__ __align__(16) float lds_pre[LL * CC];   // 8192 B: pre[j][c]

    const int tid = threadIdx.x;
    const int blk = blockIdx.x;      // blk = b*64 + i
    const int b   = blk >> 6;
    const int i   = blk & 63;

    // ---- TDM: DMA t2[b] (64x32 f32, 8 KB) into LDS with +2-dword row padding.
    // Issued by wave 0 only (EXEC is ignored by tensor ops; branch is
    // wave-uniform so other waves skip the instruction entirely).
    if (tid < 32) {
        const uint64_t ga = (uint64_t)(uintptr_t)(t2 + (size_t)b * (LL * CC));
        const uint32_t la = (uint32_t)(uintptr_t)(&lds_t2[0]); // low 32b = LDS byte offset
        // D# group 0 (CDNA5 ISA)
        v4u g0 = { 1u,                                   // count=1 (valid), no gather
                   la,                                   // lds_addr (bytes)
                   (uint32_t)ga,                         // global_addr[31:0]
                   (((uint32_t)(ga >> 32)) & 0x01FFFFFFu)
                       | (2u << 30) };                   // global_addr[56:32] | type=2
        // D# group 1 (§8.4): 1-D tile of 2048 f32 elements, pad 2 dwords
        // after every 32 dwords stored.
        v8i g1 = { (int)((2u << 16)        // data_size = 2 -> 4 bytes
                       | (1u << 20)        // pad_enable
                       | (4u << 22)        // pad_interval = 4 -> every 32 dwords
                       | (1u << 25)),      // pad_amount   = 1 -> 2 dwords
                   (int)(2048u << 16),     // tensor_dim0[15:0] = 2048 (bits 63:48)
                   (int)(1u << 16),        // tensor_dim0 hi = 0; tensor_dim1 = 1
                   (int)(2048u << 16),     // tile_dim0 = 2048
                   0,                      // tile_dim1 = tile_dim2 = 0 (unused)
                   2048,                   // tensor_dim0_stride (elements)
                   0, 0 };
        v4i gz4 = { 0, 0, 0, 0 };
        v8i gz8 = { 0, 0, 0, 0, 0, 0, 0, 0 };
        // 6-arg variant (clang-23 / therock-10.0 headers): extra int32x8
        // operand before the cache-policy immediate; unused groups zero.
        __builtin_amdgcn_tensor_load_to_lds(g0, g1, gz4, gz4, gz8, 0);
    }

    // ---- Prepass, overlapped with the TDM (disjoint LDS region):
    // pre[j][c] = t0[b,i,c] + t1[b,j,c].  t0/t1 are read exactly once per
    // block, fully coalesced; t1 index collapses to b*2048 + idx.
    {
        const float a0   = t0[(size_t)blk * CC + (tid & 31)];
        const float* t1b = t1 + (size_t)b * (LL * CC);
        #pragma unroll
        for (int r = 0; r < 8; ++r) {
            const int idx = tid + 256 * r;      // idx & 31 == tid & 31
            lds_pre[idx] = a0 + t1b[idx];
        }
    }

    if (tid < 32) __builtin_amdgcn_s_wait_tensorcnt(0);
    __syncthreads();

    // ---- Main loop: thread owns fixed k, sweeps 16 j values.
    const int k  = tid & 63;
    const int j0 = tid >> 6;                    // 0..3

    // t2 row -> 32 VGPRs via 16 conflict-free ds_load_b64 (8 B aligned: 34*4*k).
    float t2r[CC];
    {
        const float2* row = (const float2*)(&lds_t2[k * T2S]);
        #pragma unroll
        for (int h = 0; h < 16; ++h) {
            const float2 v = row[h];
            t2r[2 * h + 0] = v.x;
            t2r[2 * h + 1] = v.y;
        }
    }

    const size_t obase = (size_t)blk * (LL * LL) + (size_t)k;

    #pragma unroll 1
    for (int jj = 0; jj < 16; ++jj) {
        const int j = (jj << 2) + j0;

        // pre[j][*] -> 32 VGPRs: 8x ds_load_b128, same address in all lanes
        // of the wave (broadcast, conflict-free).
        float p[CC];
        {
            const float4* pr = (const float4*)(&lds_pre[j * CC]);
            #pragma unroll
            for (int h = 0; h < 8; ++h) {
                const float4 v = pr[h];
                p[4 * h + 0] = v.x; p[4 * h + 1] = v.y;
                p[4 * h + 2] = v.z; p[4 * h + 3] = v.w;
            }
        }

        // First-occurrence argmax over c (strict > keeps the lowest c on
        // ties, matching jnp.argmax).
        float best = p[0] + t2r[0];
        int   bi   = 0;
        #pragma unroll
        for (int c = 1; c < CC; ++c) {
            const float s  = p[c] + t2r[c];
            const bool  gt = s > best;
            bi   = gt ? c : bi;
            best = gt ? s : best;
        }

        const int distinct = (i != j) & (i != k) & (j != k);
        // Wave-contiguous in k -> 128 B/wave coalesced stores; output is
        // streaming (8 MB written once) -> nontemporal.
        __builtin_nontemporal_store((float)(bi * distinct),
                                    out + obase + (size_t)j * LL);
    }
}

extern "C" void kernel_launch(void* const* d_in, const int* in_sizes, int n_in,
                              void* d_out, int out_size, void* d_ws, size_t ws_size,
                              hipStream_t stream) {
    (void)in_sizes; (void)n_in; (void)d_ws; (void)ws_size; (void)out_size;
    const float* t0 = (const float*)d_in[0];
    const float* t1 = (const float*)d_in[1];
    const float* t2 = (const float*)d_in[2];
    float* out = (float*)d_out;

    // One block per (b, i): 512 blocks x 256 threads (8 wave32s).
    dim3 grid(BB * LL), block(256);
    hipLaunchKernelGGL(jk_argmax_kernel, grid, block, 0, stream, t0, t1, t2, out);
}